// VectorQuantiser_36412732735590
// MI455X (gfx1250) — compile-verified
//
#include <hip/hip_runtime.h>
#include <hip/hip_bf16.h>

// ---------------------------------------------------------------------------
// VectorQuantiser on gfx1250 (MI455X).
// Distance argmin via V_WMMA_F32_16X16X4_F32 (fp32 matrix cores, wave32),
// with double-buffered GLOBAL_LOAD_ASYNC_TO_LDS_B128 codebook staging
// (ASYNCcnt) overlapping the L2 codebook stream with WMMA compute.
// score(n,k) = c2[k] - 2 * <h_n, c_k>   (h^2 term dropped: constant per row)
// z_q == codebook[idx] numerically; commit_loss == codebook_loss numerically.
// ---------------------------------------------------------------------------

typedef __attribute__((ext_vector_type(2))) float v2f;
typedef __attribute__((ext_vector_type(8))) float v8f;
typedef int vq_v4i __attribute__((vector_size(16)));   // matches builtin param type

#define VQ_N       32768   // B*T
#define VQ_K       8192    // codebook size
#define VQ_D       256     // dim
#define ROWS_PER_WG 128    // h rows per workgroup (8 waves x 16 rows)
#define LDS_STRIDE  260    // 260 mod 64 == 4 -> bank-conflict-free b64 row fanout
#define NTILES     (VQ_K / 16)

// ---------------------------------------------------------------------------
// Async global->LDS helpers (CDNA5 GLOBAL_LOAD_ASYNC_TO_LDS_B128, ASYNCcnt)
// ---------------------------------------------------------------------------
__device__ __forceinline__ void async_ld_b128(const float* g, float* l) {
#if __has_builtin(__builtin_amdgcn_global_load_async_to_lds_b128)
    __builtin_amdgcn_global_load_async_to_lds_b128(
        (__attribute__((address_space(1))) vq_v4i*)(size_t)g,  // AS1: flat == global
        (__attribute__((address_space(3))) vq_v4i*)l,          // addrspacecast -> LDS offset
        0, 0);
#else
    unsigned lds_off = (unsigned)(size_t)(__attribute__((address_space(3))) void*)l;
    asm volatile("global_load_async_to_lds_b128 %0, %1, off"
                 :: "v"(lds_off), "v"((unsigned long long)(size_t)g)
                 : "memory");
#endif
}

__device__ __forceinline__ void wait_async0() {
#if __has_builtin(__builtin_amdgcn_s_wait_asynccnt)
    __builtin_amdgcn_s_wait_asynccnt(0);
#else
    asm volatile("s_wait_asynccnt 0x0" ::: "memory");
#endif
}

// ---------------------------------------------------------------------------
// Kernel 1: zero the loss accumulators (sumsq, mask count)
// ---------------------------------------------------------------------------
__global__ void vq_init_kernel(float* accum) {
    if (threadIdx.x < 2) accum[threadIdx.x] = 0.0f;
}

// ---------------------------------------------------------------------------
// Kernel 2: c2[k] = ||codebook[k]||^2
// ---------------------------------------------------------------------------
__global__ void vq_c2_kernel(const float* __restrict__ cb, float* __restrict__ c2) {
    int k = blockIdx.x * blockDim.x + threadIdx.x;
    if (k >= VQ_K) return;
    const float4* row = (const float4*)(cb + (size_t)k * VQ_D);
    float s = 0.0f;
#pragma unroll 4
    for (int i = 0; i < VQ_D / 4; ++i) {
        float4 v = row[i];
        s += v.x * v.x + v.y * v.y + v.z * v.z + v.w * v.w;
    }
    c2[k] = s;
}

// ---------------------------------------------------------------------------
// Kernel 3: WMMA distance argmin with async double-buffered codebook tiles.
// Block: 256 threads = 8 waves; each wave owns a 16-row strip of h.
// A fragment (16x4 f32): lane lo holds row M=lo; half hi supplies K = 2*hi+{0,1}.
// B fragment (4x16 f32): lane lo holds col N=lo; half hi supplies K = 2*hi+{0,1}.
// C/D (16x16 f32, 8 VGPRs): element (M = j + 8*hi, N = lo) in acc[j].
// ---------------------------------------------------------------------------
__global__ void __launch_bounds__(256)
vq_argmin_kernel(const float* __restrict__ h,
                 const float* __restrict__ cb,
                 const float* __restrict__ c2,
                 int*   __restrict__ idx_ws,
                 float* __restrict__ idx_out) {
    extern __shared__ float smem[];
    float* sh_h  = smem;                                   // ROWS_PER_WG x LDS_STRIDE
    float* sh_c0 = smem + ROWS_PER_WG * LDS_STRIDE;        // 16 x LDS_STRIDE
    float* sh_c1 = sh_c0 + 16 * LDS_STRIDE;                // 16 x LDS_STRIDE

    const int tid  = threadIdx.x;
    const int wave = tid >> 5;
    const int lane = tid & 31;
    const int lo   = lane & 15;
    const int hi   = lane >> 4;
    const int row0 = blockIdx.x * ROWS_PER_WG;
    const int wrow = wave * 16;  // wave's row strip inside the block

    // Per-thread staging coordinates for 16x256 codebook tiles:
    // thread covers (r0 + 4*it, c4) for it = 0..3, one b128 each.
    const int r0 = tid >> 6;      // 0..3
    const int c4 = tid & 63;      // float4 column

    // Stage the block's 128x256 h tile into LDS (coalesced float4, one-time).
    for (int i = tid; i < ROWS_PER_WG * (VQ_D / 4); i += 256) {
        int r = i >> 6;
        int c = i & 63;
        float4 v = ((const float4*)(h + (size_t)(row0 + r) * VQ_D))[c];
        *(float4*)&sh_h[r * LDS_STRIDE + c * 4] = v;
    }

    // Prologue: async-stage codebook tile 0 into buffer 0.
    {
        const float* g = cb + (size_t)r0 * VQ_D + c4 * 4;
        float*       l = sh_c0 + r0 * LDS_STRIDE + c4 * 4;
#pragma unroll
        for (int it = 0; it < 4; ++it)
            async_ld_b128(g + it * 4 * VQ_D, l + it * 4 * LDS_STRIDE);
    }
    wait_async0();
    __syncthreads();

    float best[8];
    int   bidx[8];
#pragma unroll
    for (int j = 0; j < 8; ++j) { best[j] = 3.4e38f; bidx[j] = 0; }

    const float* ap = &sh_h[(wrow + lo) * LDS_STRIDE + 2 * hi];

    for (int ti = 0; ti < NTILES; ++ti) {
        const float* cur = (ti & 1) ? sh_c1 : sh_c0;

        // Prefetch next tile into the other buffer (overlaps with WMMA below).
        if (ti + 1 < NTILES) {
            float*       nxt = (ti & 1) ? sh_c0 : sh_c1;
            const float* g   = cb + (size_t)(16 * (ti + 1) + r0) * VQ_D + c4 * 4;
            float*       l   = nxt + r0 * LDS_STRIDE + c4 * 4;
#pragma unroll
            for (int it = 0; it < 4; ++it)
                async_ld_b128(g + it * 4 * VQ_D, l + it * 4 * LDS_STRIDE);
        }

        const float* bp = &cur[lo * LDS_STRIDE + 2 * hi];
        v8f acc = {};
#pragma unroll
        for (int kk = 0; kk < VQ_D / 4; ++kk) {   // 64 WMMAs over the D=256 reduction
            v2f a = *(const v2f*)(ap + 4 * kk);
            v2f b = *(const v2f*)(bp + 4 * kk);
            acc = __builtin_amdgcn_wmma_f32_16x16x4_f32(
                /*neg_a=*/false, a, /*neg_b=*/false, b,
                /*c_mod=*/(short)0, acc, /*reuse_a=*/false, /*reuse_b=*/false);
        }

        // score = c2[col] - 2*dot ; track running (min, argmin) per output row.
        const int   col = 16 * ti + lo;
        const float c2v = c2[col];
#pragma unroll
        for (int j = 0; j < 8; ++j) {
            float s = __builtin_fmaf(-2.0f, acc[j], c2v);
            if (s < best[j]) { best[j] = s; bidx[j] = col; }
        }

        wait_async0();     // this wave's prefetch landed in LDS
        __syncthreads();   // all waves' prefetches visible; safe to swap buffers
    }

    // Reduce (min, argmin) across the 16 lanes of each half (N dimension).
#pragma unroll
    for (int j = 0; j < 8; ++j) {
        for (int m = 8; m >= 1; m >>= 1) {
            float ob = __shfl_xor(best[j], m, 32);
            int   oi = __shfl_xor(bidx[j], m, 32);
            if (ob < best[j] || (ob == best[j] && oi < bidx[j])) {
                best[j] = ob; bidx[j] = oi;
            }
        }
        if (lo == 0) {
            int row = row0 + wrow + j + 8 * hi;   // C layout: M = j + 8*hi
            idx_ws[row]  = bidx[j];
            idx_out[row] = (float)bidx[j];
        }
    }
}

// ---------------------------------------------------------------------------
// Kernel 4: gather z_q = codebook[idx], masked squared error + mask count.
// One wave per h row; one atomicAdd pair per block.
// ---------------------------------------------------------------------------
__global__ void __launch_bounds__(256)
vq_gather_loss_kernel(const float* __restrict__ h,
                      const int*   __restrict__ mask,
                      const float* __restrict__ cb,
                      const int*   __restrict__ idx_ws,
                      float* __restrict__ zq,
                      float* __restrict__ accum) {
    __shared__ float bs[8];
    __shared__ int   bc[8];
    const int wave = threadIdx.x >> 5;
    const int lane = threadIdx.x & 31;
    const int n    = blockIdx.x * 8 + wave;

    const int id = idx_ws[n];
    const int m  = mask[n];
    const float4* hr = (const float4*)(h  + (size_t)n  * VQ_D);
    const float4* zr = (const float4*)(cb + (size_t)id * VQ_D);
    float4*      out = (float4*)(zq + (size_t)n * VQ_D);

    float s = 0.0f;
#pragma unroll
    for (int i = lane; i < VQ_D / 4; i += 32) {
        float4 hv = hr[i];
        float4 zv = zr[i];
        out[i] = zv;
        if (m) {
            float dx = hv.x - zv.x, dy = hv.y - zv.y;
            float dz = hv.z - zv.z, dw = hv.w - zv.w;
            s += dx * dx + dy * dy + dz * dz + dw * dw;
        }
    }
    for (int mm = 16; mm >= 1; mm >>= 1) s += __shfl_xor(s, mm, 32);
    if (lane == 0) { bs[wave] = s; bc[wave] = (m != 0) ? 1 : 0; }
    __syncthreads();
    if (threadIdx.x == 0) {
        float t = 0.0f; int c = 0;
        for (int i = 0; i < 8; ++i) { t += bs[i]; c += bc[i]; }
        atomicAdd(&accum[0], t);
        atomicAdd(&accum[1], (float)c);
    }
}

// ---------------------------------------------------------------------------
// Kernel 5: finalize both losses (numerically identical values).
// ---------------------------------------------------------------------------
__global__ void vq_finalize_kernel(const float* __restrict__ accum,
                                   float* __restrict__ loss_out) {
    if (threadIdx.x == 0) {
        float denom = accum[1] * (float)VQ_D + 1e-8f;
        float loss  = accum[0] / denom;
        loss_out[0] = loss;   // commit_loss
        loss_out[1] = loss;   // codebook_loss
    }
}

// ---------------------------------------------------------------------------
// Host launcher
// ---------------------------------------------------------------------------
extern "C" void kernel_launch(void* const* d_in, const int* in_sizes, int n_in,
                              void* d_out, int out_size, void* d_ws, size_t ws_size,
                              hipStream_t stream) {
    const float* h    = (const float*)d_in[0];   // (8,4096,256) f32
    const int*   mask = (const int*)d_in[1];     // (8,4096) bool->int
    const float* cb   = (const float*)d_in[2];   // (8192,256) f32

    // Workspace layout (floats): [c2: 8192][idx(int): 32768][accum: 2]
    float* ws     = (float*)d_ws;
    float* c2     = ws;
    int*   idx_ws = (int*)(ws + VQ_K);
    float* accum  = ws + VQ_K + VQ_N;

    // Output layout: [z_q: N*D][indices(as float): N][commit][codebook]
    float* zq       = (float*)d_out;
    float* idx_out  = zq + (size_t)VQ_N * VQ_D;
    float* loss_out = idx_out + VQ_N;

    vq_init_kernel<<<1, 32, 0, stream>>>(accum);
    vq_c2_kernel<<<VQ_K / 256, 256, 0, stream>>>(cb, c2);

    const size_t lds_bytes =
        (size_t)(ROWS_PER_WG * LDS_STRIDE + 2 * 16 * LDS_STRIDE) * sizeof(float);
    vq_argmin_kernel<<<VQ_N / ROWS_PER_WG, 256, lds_bytes, stream>>>(
        h, cb, c2, idx_ws, idx_out);

    vq_gather_loss_kernel<<<VQ_N / 8, 256, 0, stream>>>(
        h, mask, cb, idx_ws, zq, accum);

    vq_finalize_kernel<<<1, 32, 0, stream>>>(accum, loss_out);
}